// kernel_manual_0_21887153341178
// MI455X (gfx1250) — compile-verified
//
#include <hip/hip_runtime.h>

typedef __attribute__((ext_vector_type(2))) float v2f;
typedef __attribute__((ext_vector_type(8))) float v8f;
typedef __attribute__((ext_vector_type(4))) unsigned int v4u;
typedef __attribute__((ext_vector_type(4))) int v4i;
typedef __attribute__((ext_vector_type(8))) int v8i;

#define IMGW  56
#define HWSZ  3136          // 56*56
#define PW    58            // padded width
#define PHW   3364          // 58*58 (per-channel padded plane)
#define KTOT  288           // 32 in-channels * 9 taps
#define BN    64            // output channels
#define BELEM (KTOT * BN)   // 18432 floats of reordered weights

// Kernel 1: fused depthwise 1x3 conv (W pad 1) + adjacent-channel pair sum,
// written into a zero-padded 32 x 58 x 58 buffer (1-pixel halo of zeros) so
// the WMMA GEMM needs no boundary checks at all.
__global__ void dw_pairsum_pad_kernel(const float* __restrict__ x,
                                      const float* __restrict__ wdw,
                                      float* __restrict__ t5p) {
  int idx = blockIdx.x * blockDim.x + threadIdx.x;
  if (idx >= 32 * PHW) return;
  int c  = idx / PHW;
  int r  = idx % PHW;
  int hp = r / PW;
  int wp = r % PW;
  int h  = hp - 1;
  int w  = wp - 1;
  float acc = 0.f;
  if (h >= 0 && h < IMGW && w >= 0 && w < IMGW) {
#pragma unroll
    for (int e = 0; e < 2; ++e) {
      int ch = 2 * c + e;
      const float* xr = x + ch * HWSZ + h * IMGW;
      float w0 = wdw[ch * 3 + 0];
      float w1 = wdw[ch * 3 + 1];
      float w2 = wdw[ch * 3 + 2];
      float left  = (w > 0)        ? xr[w - 1] : 0.f;
      float mid   =                  xr[w];
      float right = (w < IMGW - 1) ? xr[w + 1] : 0.f;
      acc += left * w0 + mid * w1 + right * w2;
    }
  }
  t5p[idx] = acc;
}

// Kernel 2: reorder weights (O=64, C=32, KH=3, KW=3) -> Bt[tap][c][n]
// (tap-major K order matching the GEMM), lane-coalesced in n.
__global__ void transpose_w_kernel(const float* __restrict__ wconv,
                                   float* __restrict__ Bt) {
  int idx = blockIdx.x * blockDim.x + threadIdx.x;
  if (idx >= BELEM) return;
  int n   = idx % BN;
  int kt  = idx / BN;       // tap*32 + c
  int tap = kt / 32;
  int c   = kt % 32;
  Bt[idx] = wconv[n * KTOT + c * 9 + tap];
}

// Kernel 3: implicit-GEMM 3x3 conv via V_WMMA_F32_16X16X4_F32.
// Weights are DMA'd into LDS once per block with the Tensor Data Mover
// (tensor_load_to_lds, TENSORcnt), then read via ds_load.
// M = 3136 pixels (16-pixel tiles), N = 64 channels (16-chan tiles),
// K = 9 taps x 32 channels, fully unrolled -> 72 WMMAs.
// Block: 256 threads = 8 waves = 2 m-tiles x 4 n-tiles. Grid: 98 blocks.
__global__ void __launch_bounds__(256)
conv3x3_wmma_kernel(const float* __restrict__ t5p,
                    const float* __restrict__ Bt,
                    float* __restrict__ out) {
  extern __shared__ float lds_B[];        // KTOT*BN floats = 73,728 B

  const int lane  = threadIdx.x & 31;     // wave32
  const int wid   = threadIdx.x >> 5;
  const int l16   = lane & 15;
  const int khalf = lane >> 4;            // 0: K={k,k+1}, 1: K={k+2,k+3}
  const int m_tile = blockIdx.x * 2 + (wid >> 2);
  const int n_tile = wid & 3;

  // ---- Stage B (weights) into LDS with the TDM, one wave per block ----
#if __has_builtin(__builtin_amdgcn_tensor_load_to_lds)
  if (threadIdx.x < 32) {                 // wave-uniform: only wave 0 issues
    unsigned long long ga = (unsigned long long)(uintptr_t)Bt;
    unsigned int lds_off  = (unsigned int)(uintptr_t)&lds_B[0];

    v4u g0;
    g0[0] = 1u;                           // count=1, is_restore=0, no gather
    g0[1] = lds_off;                      // lds_addr (bytes)
    g0[2] = (unsigned int)ga;             // global_addr[31:0]
    g0[3] = (unsigned int)(ga >> 32) | (2u << 30);  // addr[56:32] | type=2

    v8i g1;
    g1[0] = (2 << 16);                    // wg_mask=0, data_size=2 (4 bytes)
    g1[1] = (int)((BELEM & 0xFFFF) << 16);        // tensor_dim0[15:0] @ b63:48
    g1[2] = (int)(((BELEM >> 16) & 0xFFFF) | (1 << 16)); // dim0 hi, tensor_dim1=1
    g1[3] = (int)((BELEM & 0xFFFF) << 16);        // tile_dim0 = 18432 @ b127:112
    g1[4] = 1;                            // tile_dim1 = 1, tile_dim2 = 0
    g1[5] = BELEM;                        // tensor_dim0_stride[31:0]
    g1[6] = 0;                            // stride0 hi, tensor_dim1_stride lo
    g1[7] = 0;

    v4i g2 = {0, 0, 0, 0};                // unused (<=2D tensor)
    v4i g3 = {0, 0, 0, 0};
    v8i g4 = {0, 0, 0, 0, 0, 0, 0, 0};    // unused trailing group (6-arg form)

    __builtin_amdgcn_tensor_load_to_lds(g0, g1, g2, g3, g4, 0);
    __builtin_amdgcn_s_wait_tensorcnt(0);
  }
#else
  for (int i = threadIdx.x; i < BELEM; i += 256) lds_B[i] = Bt[i];
#endif
  __syncthreads();

  // A-fragment row (pixel) owned by this lane, in the padded image
  const int p  = m_tile * 16 + l16;
  const int h  = p / IMGW;
  const int w  = p % IMGW;
  const int a0 = h * PW + w + khalf * 2 * PHW;  // tap (0,0) base for this lane
  // B-fragment column (output channel) owned by this lane
  const int n  = n_tile * 16 + l16;

  const float* __restrict__ abase = t5p + a0;
  const float* bbase = lds_B + khalf * 2 * BN + n;

  v8f acc = {};
#pragma unroll
  for (int kh = 0; kh < 3; ++kh) {
#pragma unroll
    for (int kw = 0; kw < 3; ++kw) {
      const float* __restrict__ ap = abase + kh * PW + kw;            // + const
      const float* bp = bbase + (kh * 3 + kw) * 32 * BN;              // + const
#pragma unroll
      for (int c4 = 0; c4 < 8; ++c4) {
        v2f a, b;
        a[0] = ap[(c4 * 4 + 0) * PHW];   // immediate-offset global loads
        a[1] = ap[(c4 * 4 + 1) * PHW];
        b[0] = bp[(c4 * 4 + 0) * BN];    // ds_load_b32, immediate offsets
        b[1] = bp[(c4 * 4 + 1) * BN];
        acc = __builtin_amdgcn_wmma_f32_16x16x4_f32(
            /*neg_a=*/false, a, /*neg_b=*/false, b,
            /*c_mod=*/(short)0, acc, /*reuse_a=*/false, /*reuse_b=*/false);
      }
    }
  }

  // C/D layout: lanes 0-15 -> M = v (VGPR v), lanes 16-31 -> M = v + 8
#pragma unroll
  for (int v = 0; v < 8; ++v) {
    const int m  = v + khalf * 8;
    const int pp = m_tile * 16 + m;
    out[n * HWSZ + pp] = acc[v];
  }
}

extern "C" void kernel_launch(void* const* d_in, const int* in_sizes, int n_in,
                              void* d_out, int out_size, void* d_ws, size_t ws_size,
                              hipStream_t stream) {
  const float* x     = (const float*)d_in[0];  // (1,64,56,56)
  const float* wconv = (const float*)d_in[1];  // (64,32,3,3,1)
  const float* wdw   = (const float*)d_in[2];  // (64,3,1)
  float* out = (float*)d_out;                  // (1,64,56,56)

  float* t5p = (float*)d_ws;                   // 32*58*58 floats (zero-padded)
  float* Bt  = t5p + 32 * PHW;                 // 288*64 floats

  dw_pairsum_pad_kernel<<<(32 * PHW + 255) / 256, 256, 0, stream>>>(x, wdw, t5p);
  transpose_w_kernel<<<(BELEM + 255) / 256, 256, 0, stream>>>(wconv, Bt);
  conv3x3_wmma_kernel<<<98, 256, (size_t)(BELEM * sizeof(float)), stream>>>(
      t5p, Bt, out);
}